// TriplanarFeatureVolume_37812892074356
// MI455X (gfx1250) — compile-verified
//
#include <hip/hip_runtime.h>
#include <hip/hip_bf16.h>

// Problem constants (from reference):
// BATCH=4, NSAMP=262144, FDIM=32, FSIZE=512 -> planes are [32, 513, 513] f32.
#define FDIM        32
#define FS          513
#define PLANE_HW    (FS * FS)            // 263169
#define PLANE_ELEMS ((size_t)FDIM * PLANE_HW)
#define NS_TOTAL    (4 * 262144)         // 1,048,576 samples total
#define SPAN        512.0f
#define TWOSPAN     1024.0f
#define X0MAX       (FS - 2)             // 511

// Reflection padding, align_corners=true (matches reference _reflect()).
__device__ __forceinline__ float reflectf(float v) {
    v = fabsf(v);
    v = v - TWOSPAN * floorf(v * (1.0f / TWOSPAN));   // exact: /1024 is a pow2 scale
    v = (v > SPAN) ? (TWOSPAN - v) : v;
    return fminf(fmaxf(v, 0.0f), SPAN);
}

// --------------------------------------------------------------------------
// Pass 1: transpose each plane [C,H,W] -> [H,W,C] into workspace so that the
// 32 channels of any (y,x) texel are one contiguous 128B cache line.
// Block = 256 threads, tile = 256 x-values for one y, one plane.
// LDS tile padded to 33 floats/row -> conflict-free on both phases.
// --------------------------------------------------------------------------
__global__ void __launch_bounds__(256)
triplanar_transpose_kernel(const float* __restrict__ fmx,
                           const float* __restrict__ fmy,
                           const float* __restrict__ fmz,
                           float* __restrict__ ws) {
    __shared__ float tile[256 * 33];

    const int tid = threadIdx.x;
    const int x0  = blockIdx.x * 256;
    const int y   = blockIdx.y;
    const int p   = blockIdx.z;

    const float* __restrict__ src = (p == 0) ? fmx : ((p == 1) ? fmy : fmz);
    float* __restrict__ dst = ws + (size_t)p * PLANE_ELEMS;

    const int xg = x0 + tid;
    if (xg < FS) {
        const float* s = src + (size_t)y * FS + xg;
        #pragma unroll
        for (int c = 0; c < FDIM; ++c) {
            // source plane is read exactly once by this pass -> non-temporal
            tile[tid * 33 + c] = __builtin_nontemporal_load(s + (size_t)c * PLANE_HW);
        }
    }
    __syncthreads();

    const int nvalid = min(256, FS - x0);                 // valid x in this tile
    float* dbase = dst + ((size_t)y * FS + x0) * FDIM;    // [H,W,C] destination
    #pragma unroll
    for (int pass = 0; pass < 32; ++pass) {
        const int f  = pass * 256 + tid;                  // flat (xl, c) index
        const int xl = f >> 5;
        const int c  = f & 31;
        if (xl < nvalid) {
            // want these resident in L2 for the gather pass -> regular (RT) store
            dbase[f] = tile[xl * 33 + c];
        }
    }
}

// --------------------------------------------------------------------------
// Pass 2: gather. One wave32 per sample, one lane per channel.
// Transposed layout: each corner fetch = one coalesced 128B line; each
// per-plane output = one coalesced 128B NT store (don't pollute L2).
// --------------------------------------------------------------------------
__device__ __forceinline__ void sample_plane_t(const float* __restrict__ pl,
                                               float gx, float gy, int lane,
                                               float* __restrict__ outp) {
    const float ix = reflectf(fmaf(gx, 256.0f, 256.0f));
    const float iy = reflectf(fmaf(gy, 256.0f, 256.0f));
    // Clamp base texel to <= W-2 so the +1 neighbor is always in-bounds;
    // wx/wy then reach 1.0 exactly at the far edge (same result as reference).
    const int   x0 = min((int)floorf(ix), X0MAX);
    const int   y0 = min((int)floorf(iy), X0MAX);
    const float wx = ix - (float)x0;
    const float wy = iy - (float)y0;

    const size_t base = ((size_t)y0 * FS + x0) * FDIM + lane;
    const float v00 = pl[base];
    const float v01 = pl[base + FDIM];
    const float v10 = pl[base + (size_t)FS * FDIM];
    const float v11 = pl[base + (size_t)FS * FDIM + FDIM];

    const float r0 = fmaf(v01 - v00, wx, v00);
    const float r1 = fmaf(v11 - v10, wx, v10);
    const float r  = fmaf(r1 - r0, wy, r0);
    __builtin_nontemporal_store(r, outp + lane);
}

__global__ void __launch_bounds__(256)
triplanar_gather_t_kernel(const float* __restrict__ xin,
                          const float* __restrict__ wsT,
                          float* __restrict__ out) {
    const int lane = threadIdx.x & 31;
    const int gs   = blockIdx.x * (blockDim.x >> 5) + (threadIdx.x >> 5);
    if (gs >= NS_TOTAL) return;

    const float* xp = xin + (size_t)gs * 3;
    const float g0 = __builtin_nontemporal_load(xp + 0);
    const float g1 = __builtin_nontemporal_load(xp + 1);
    const float g2 = __builtin_nontemporal_load(xp + 2);

    float* obase = out + (size_t)gs * 3 * FDIM;
    const float* p0 = wsT;
    const float* p1 = wsT + PLANE_ELEMS;
    const float* p2 = wsT + 2 * PLANE_ELEMS;

    sample_plane_t(p0, g1, g2, lane, obase);             // sx: (x[...,1], x[...,2]) on fmx
    sample_plane_t(p1, g0, g2, lane, obase + FDIM);      // sy: (x[...,0], x[...,2]) on fmy
    sample_plane_t(p2, g0, g1, lane, obase + 2 * FDIM);  // sz: (x[...,0], x[...,1]) on fmz
}

// --------------------------------------------------------------------------
// Fallback (only if ws_size < 97MB): gather straight from [C,H,W] layout.
// Lane-per-channel, channel stride PLANE_HW. Correct but uncoalesced.
// --------------------------------------------------------------------------
__device__ __forceinline__ void sample_plane_chw(const float* __restrict__ pl,
                                                 float gx, float gy, int lane,
                                                 float* __restrict__ outp) {
    const float ix = reflectf(fmaf(gx, 256.0f, 256.0f));
    const float iy = reflectf(fmaf(gy, 256.0f, 256.0f));
    const int   x0 = min((int)floorf(ix), X0MAX);
    const int   y0 = min((int)floorf(iy), X0MAX);
    const float wx = ix - (float)x0;
    const float wy = iy - (float)y0;

    const size_t base = (size_t)lane * PLANE_HW + (size_t)y0 * FS + x0;
    const float v00 = pl[base];
    const float v01 = pl[base + 1];
    const float v10 = pl[base + FS];
    const float v11 = pl[base + FS + 1];

    const float r0 = fmaf(v01 - v00, wx, v00);
    const float r1 = fmaf(v11 - v10, wx, v10);
    const float r  = fmaf(r1 - r0, wy, r0);
    __builtin_nontemporal_store(r, outp + lane);
}

__global__ void __launch_bounds__(256)
triplanar_gather_chw_kernel(const float* __restrict__ xin,
                            const float* __restrict__ fmx,
                            const float* __restrict__ fmy,
                            const float* __restrict__ fmz,
                            float* __restrict__ out) {
    const int lane = threadIdx.x & 31;
    const int gs   = blockIdx.x * (blockDim.x >> 5) + (threadIdx.x >> 5);
    if (gs >= NS_TOTAL) return;

    const float* xp = xin + (size_t)gs * 3;
    const float g0 = __builtin_nontemporal_load(xp + 0);
    const float g1 = __builtin_nontemporal_load(xp + 1);
    const float g2 = __builtin_nontemporal_load(xp + 2);

    float* obase = out + (size_t)gs * 3 * FDIM;
    sample_plane_chw(fmx, g1, g2, lane, obase);
    sample_plane_chw(fmy, g0, g2, lane, obase + FDIM);
    sample_plane_chw(fmz, g0, g1, lane, obase + 2 * FDIM);
}

extern "C" void kernel_launch(void* const* d_in, const int* in_sizes, int n_in,
                              void* d_out, int out_size, void* d_ws, size_t ws_size,
                              hipStream_t stream) {
    (void)in_sizes; (void)n_in; (void)out_size;
    const float* x   = (const float*)d_in[0];
    const float* fmx = (const float*)d_in[1];
    const float* fmy = (const float*)d_in[2];
    const float* fmz = (const float*)d_in[3];
    float* out = (float*)d_out;

    const size_t need = 3 * PLANE_ELEMS * sizeof(float);   // ~96.4 MiB
    const int gather_blocks = NS_TOTAL / 8;                 // 8 waves/block, 1 sample/wave

    if (ws_size >= need) {
        float* wsT = (float*)d_ws;
        dim3 tg((FS + 255) / 256, FS, 3);                  // 3 x-tiles, 513 y, 3 planes
        triplanar_transpose_kernel<<<tg, 256, 0, stream>>>(fmx, fmy, fmz, wsT);
        triplanar_gather_t_kernel<<<gather_blocks, 256, 0, stream>>>(x, wsT, out);
    } else {
        triplanar_gather_chw_kernel<<<gather_blocks, 256, 0, stream>>>(x, fmx, fmy, fmz, out);
    }
}